// RealMambaPredictor_82523501625835
// MI455X (gfx1250) — compile-verified
//
#include <hip/hip_runtime.h>
#include <hip/hip_bf16.h>

typedef __attribute__((ext_vector_type(2))) float v2f;
typedef __attribute__((ext_vector_type(4))) float v4f;
typedef __attribute__((ext_vector_type(8))) float v8f;

#define DEV __device__ __forceinline__

constexpr int kS     = 2048;
constexpr int kBATCH = 2048;

#if __has_builtin(__builtin_amdgcn_tanhf)
#define TANHF(x) __builtin_amdgcn_tanhf(x)     // native V_TANH_F32 on gfx1250
#define HAVE_NATIVE_TANH 1
#elif __has_builtin(__builtin_amdgcn_tanh_f32)
#define TANHF(x) __builtin_amdgcn_tanh_f32(x)
#define HAVE_NATIVE_TANH 1
#else
#define TANHF(x) tanhf(x)
#define HAVE_NATIVE_TANH 0
#endif

DEV float sigmoidf_fast(float x) {
#if HAVE_NATIVE_TANH
  // sigmoid(x) == 0.5 + 0.5*tanh(x/2) exactly; 3 ops on the TRANS/VALU pipes
  return __builtin_fmaf(0.5f, TANHF(0.5f * x), 0.5f);
#else
  float e = __builtin_amdgcn_exp2f(x * -1.44269504f);
  return __builtin_amdgcn_rcpf(1.f + e);
#endif
}

// ---- 16x16x16 fp32 matmul as 4 chained V_WMMA_F32_16X16X4_F32 ----
DEV v8f mm16(const v2f W[4], const v2f X[4], v8f acc) {
#pragma unroll
  for (int c = 0; c < 4; ++c)
    acc = __builtin_amdgcn_wmma_f32_16x16x4_f32(
        false, W[c], false, X[c], (short)0, acc, false, false);
  return acc;
}

// Convert 16x16 (feat x batch) from C/D layout into B-operand chunks.
// C/D: VGPR r holds feat r (lanes 0-15) / feat r+8 (lanes 16-31), batch=lane&15.
// B chunk c, vgpr j, lane l: K = 4c + j + 2*(l>>4), N = l&15.
#if __has_builtin(__builtin_amdgcn_permlane16_swap)
DEV void swap16(float& a, float& c) {
  // v_permlane16_swap_b32: a' = {a.row0|c.row0}, c' = {a.row1|c.row1}
  auto r = __builtin_amdgcn_permlane16_swap(__float_as_uint(a),
                                            __float_as_uint(c), false, false);
  a = __uint_as_float(r[0]);
  c = __uint_as_float(r[1]);
}
DEV void d_to_b(v8f d, v2f b[4], int lane) {
  (void)lane;
  float a0 = d[0], c0 = d[2]; swap16(a0, c0); b[0].x = a0; b[2].x = c0;
  float a1 = d[1], c1 = d[3]; swap16(a1, c1); b[0].y = a1; b[2].y = c1;
  float a2 = d[4], c2 = d[6]; swap16(a2, c2); b[1].x = a2; b[3].x = c2;
  float a3 = d[5], c3 = d[7]; swap16(a3, c3); b[1].y = a3; b[3].y = c3;
}
#else
DEV void d_to_b(v8f d, v2f b[4], int lane) {
  const int idlo = lane & 15;
  const int idhi = (lane & 15) + 16;
  const bool lo = lane < 16;
  float u2 = __shfl(d[2], idlo, 32), u3 = __shfl(d[3], idlo, 32);
  float u6 = __shfl(d[6], idlo, 32), u7 = __shfl(d[7], idlo, 32);
  float w0 = __shfl(d[0], idhi, 32), w1 = __shfl(d[1], idhi, 32);
  float w4 = __shfl(d[4], idhi, 32), w5 = __shfl(d[5], idhi, 32);
  b[0].x = lo ? d[0] : u2;  b[0].y = lo ? d[1] : u3;
  b[1].x = lo ? d[4] : u6;  b[1].y = lo ? d[5] : u7;
  b[2].x = lo ? w0 : d[2];  b[2].y = lo ? w1 : d[3];
  b[3].x = lo ? w4 : d[6];  b[3].y = lo ? w5 : d[7];
}
#endif

// Load one D-layout constant vector (8 contiguous floats) from LDS.
DEV v8f ldc(const float* cl, int vec, int f8) {
  const v4f* p = reinterpret_cast<const v4f*>(cl + vec * 16 + f8);
  v4f a = p[0], b = p[1];
  v8f r;
  r[0] = a[0]; r[1] = a[1]; r[2] = a[2]; r[3] = a[3];
  r[4] = b[0]; r[5] = b[1]; r[6] = b[2]; r[7] = b[3];
  return r;
}

__global__ __launch_bounds__(128)
void mamba_fused(const float* __restrict__ x,
                 const float* __restrict__ emb_W, const float* __restrict__ emb_b,
                 const float* __restrict__ in_W,  const float* __restrict__ in_b,
                 const float* __restrict__ Ap,    const float* __restrict__ Bp,
                 const float* __restrict__ Cp,    const float* __restrict__ Dpp,
                 const float* __restrict__ out_W, const float* __restrict__ out_b,
                 const float* __restrict__ ln_g,  const float* __restrict__ ln_b,
                 const float* __restrict__ pW1,   const float* __restrict__ pb1,
                 const float* __restrict__ pW2,   const float* __restrict__ pb2,
                 const float* __restrict__ uW1,   const float* __restrict__ ub1,
                 const float* __restrict__ uW2,   const float* __restrict__ ub2,
                 float* __restrict__ out)
{
  __shared__ float cl[13 * 16];     // D-layout constant vectors (per block)
  __shared__ float hbuf[4 * 256];   // per-wave last-hidden handoff for heads

  const int lane   = threadIdx.x & 31;
  const int wave   = threadIdx.x >> 5;
  const int batch0 = (blockIdx.x * 4 + wave) * 16;
  const int m  = lane & 15;                   // A-operand row / batch column
  const int kh = (lane >> 4) * 2;             // K offset of this lane half
  const int f8 = (lane >> 4) * 8;             // feature offset in D layout

  // ---- constant vectors -> LDS (written once per block) ----
  // vec 0: emb_b ; per layer l base 1+6l: +0 in_b[:16] +1 in_b[16:] +2 Dp +3 out_b +4 ln_g +5 ln_b
  if (threadIdx.x < 16) {
    int i = threadIdx.x;
    cl[0 * 16 + i] = emb_b[i];
#pragma unroll
    for (int l = 0; l < 2; ++l) {
      cl[(1 + 6 * l) * 16 + i] = in_b[l * 32 + i];
      cl[(2 + 6 * l) * 16 + i] = in_b[l * 32 + 16 + i];
      cl[(3 + 6 * l) * 16 + i] = Dpp[l * 16 + i];
      cl[(4 + 6 * l) * 16 + i] = out_b[l * 16 + i];
      cl[(5 + 6 * l) * 16 + i] = ln_g[l * 16 + i];
      cl[(6 + 6 * l) * 16 + i] = ln_b[l * 16 + i];
    }
  }
  __syncthreads();

  // ---------------- weight preload into A-operand registers ----------------
  v2f Wemb[4], W1[2][4], W2[2][4], Wa[2][4], Wb[2][4], Wc[2][4], Wo[2][4];
#pragma unroll
  for (int c = 0; c < 4; ++c) {
    int k0 = 4 * c + kh;
    Wemb[c].x = emb_W[k0 * 16 + m];           // emb_W^T
    Wemb[c].y = emb_W[(k0 + 1) * 16 + m];
  }
#pragma unroll
  for (int l = 0; l < 2; ++l) {
#pragma unroll
    for (int c = 0; c < 4; ++c) {
      int k0 = 4 * c + kh;
      W1[l][c].x = in_W[l * 512 + k0 * 32 + m];            // in_W[:, :16]^T
      W1[l][c].y = in_W[l * 512 + (k0 + 1) * 32 + m];
      W2[l][c].x = in_W[l * 512 + k0 * 32 + 16 + m];       // in_W[:, 16:]^T
      W2[l][c].y = in_W[l * 512 + (k0 + 1) * 32 + 16 + m];
      Wa[l][c].x = Ap[l * 256 + k0 * 16 + m];              // A^T
      Wa[l][c].y = Ap[l * 256 + (k0 + 1) * 16 + m];
      Wb[l][c].x = Bp[l * 256 + m * 16 + k0];              // B  (h@B^T)
      Wb[l][c].y = Bp[l * 256 + m * 16 + k0 + 1];
      Wc[l][c].x = Cp[l * 256 + m * 16 + k0];              // C  (h@C^T)
      Wc[l][c].y = Cp[l * 256 + m * 16 + k0 + 1];
      Wo[l][c].x = out_W[l * 256 + k0 * 16 + m];           // out_W^T
      Wo[l][c].y = out_W[l * 256 + (k0 + 1) * 16 + m];
    }
  }

  const v8f vzero = {0.f, 0.f, 0.f, 0.f, 0.f, 0.f, 0.f, 0.f};
  v2f Sb[2][4];                                // scan states, B layout
#pragma unroll
  for (int l = 0; l < 2; ++l)
#pragma unroll
    for (int c = 0; c < 4; ++c) { Sb[l][c].x = 0.f; Sb[l][c].y = 0.f; }

  const float* xrow = x + (size_t)(batch0 + m) * kS * 16;
  v8f h = vzero;

  // ---------------- stream the sequence through both layers ----------------
#pragma unroll 1
  for (int t = 0; t < kS; ++t) {
    __syncthreads();   // keeps const reloads in-loop (reg pressure) + waves in step
    const float* xt = xrow + (size_t)t * 16;
    __builtin_prefetch(xt + 16 * 8, 0, 3);     // WGP-scope stream-ahead prefetch
    v2f Xb[4];
#pragma unroll
    for (int c = 0; c < 4; ++c)
      Xb[c] = *reinterpret_cast<const v2f*>(xt + 4 * c + kh);

    h = mm16(Wemb, Xb, ldc(cl, 0, f8));        // embedding -> D layout

#pragma unroll
    for (int l = 0; l < 2; ++l) {
      const int cb = 1 + 6 * l;
      v8f res = h;
      v2f Hb[4]; d_to_b(h, Hb, lane);
      v8f x1 = mm16(W1[l], Hb, ldc(cl, cb + 0, f8));
      v8f x2 = mm16(W2[l], Hb, ldc(cl, cb + 1, f8));
      v2f X1b[4]; d_to_b(x1, X1b, lane);

      // s_new = tanh(x1@A + s_old@B^T)
      v8f pre = mm16(Wa[l], X1b, vzero);
      pre = mm16(Wb[l], Sb[l], pre);
      v8f s;
#pragma unroll
      for (int r = 0; r < 8; ++r) s[r] = TANHF(pre[r]);
      d_to_b(s, Sb[l], lane);                  // reused next step as s_old

      // y = (s_new@C^T + x1*Dp) * sigmoid(x2)
      v8f y  = mm16(Wc[l], Sb[l], vzero);
      v8f dp = ldc(cl, cb + 2, f8);
#pragma unroll
      for (int r = 0; r < 8; ++r)
        y[r] = (y[r] + x1[r] * dp[r]) * sigmoidf_fast(x2[r]);

      v2f Yb[4]; d_to_b(y, Yb, lane);
      v8f o = mm16(Wo[l], Yb, ldc(cl, cb + 3, f8));

      // h = layernorm(o + res): 16 feats split across the two lane halves
      float ssum = 0.f;
#pragma unroll
      for (int r = 0; r < 8; ++r) { o[r] += res[r]; ssum += o[r]; }
      ssum += __shfl_xor(ssum, 16, 32);
      float mean = ssum * (1.f / 16.f);
      float vsum = 0.f;
#pragma unroll
      for (int r = 0; r < 8; ++r) { float dv = o[r] - mean; vsum += dv * dv; }
      vsum += __shfl_xor(vsum, 16, 32);
      float rinv = __builtin_amdgcn_rsqf(vsum * (1.f / 16.f) + 1e-5f);
      v8f g8 = ldc(cl, cb + 4, f8);
      v8f b8 = ldc(cl, cb + 5, f8);
#pragma unroll
      for (int r = 0; r < 8; ++r)
        h[r] = (o[r] - mean) * rinv * g8[r] + b8[r];
    }
  }

  // ---------------- heads on last hidden state ----------------
#pragma unroll
  for (int r = 0; r < 8; ++r)
    hbuf[wave * 256 + m * 16 + f8 + r] = h[r];
  __syncthreads();

  if (lane < 16) {
    const float* lv = &hbuf[wave * 256 + lane * 16];
    float p = pb2[0], u = ub2[0];
#pragma unroll
    for (int j = 0; j < 8; ++j) {
      float a = pb1[j], b = ub1[j];
#pragma unroll
      for (int i = 0; i < 16; ++i) {
        a += lv[i] * pW1[i * 8 + j];
        b += lv[i] * uW1[i * 8 + j];
      }
      p += fmaxf(a, 0.f) * pW2[j];
      u += fmaxf(b, 0.f) * uW2[j];
    }
    float sp = fmaxf(u, 0.f) + log1pf(__expf(-fabsf(u)));  // stable softplus
    out[batch0 + lane] = p;
    out[kBATCH + batch0 + lane] = sp;
  }
}

extern "C" void kernel_launch(void* const* d_in, const int* in_sizes, int n_in,
                              void* d_out, int out_size, void* d_ws, size_t ws_size,
                              hipStream_t stream) {
  const float* x     = (const float*)d_in[0];
  const float* emb_W = (const float*)d_in[1];
  const float* emb_b = (const float*)d_in[2];
  const float* in_W  = (const float*)d_in[3];
  const float* in_b  = (const float*)d_in[4];
  const float* Ap    = (const float*)d_in[5];
  const float* Bp    = (const float*)d_in[6];
  const float* Cp    = (const float*)d_in[7];
  const float* Dpp   = (const float*)d_in[8];
  const float* out_W = (const float*)d_in[9];
  const float* out_b = (const float*)d_in[10];
  const float* ln_g  = (const float*)d_in[11];
  const float* ln_b  = (const float*)d_in[12];
  const float* pW1   = (const float*)d_in[13];
  const float* pb1   = (const float*)d_in[14];
  const float* pW2   = (const float*)d_in[15];
  const float* pb2   = (const float*)d_in[16];
  const float* uW1   = (const float*)d_in[17];
  const float* ub1   = (const float*)d_in[18];
  const float* uW2   = (const float*)d_in[19];
  const float* ub2   = (const float*)d_in[20];
  float* outp = (float*)d_out;

  dim3 grid(kBATCH / 64);   // 32 blocks x 4 waves x 16 batches = 2048
  dim3 block(128);
  mamba_fused<<<grid, block, 0, stream>>>(
      x, emb_W, emb_b, in_W, in_b, Ap, Bp, Cp, Dpp, out_W, out_b,
      ln_g, ln_b, pW1, pb1, pW2, pb2, uW1, ub1, uW2, ub2, outp);
}